// MultiVariateGaussian_17540646437800
// MI455X (gfx1250) — compile-verified
//
#include <hip/hip_runtime.h>

// Problem constants (fixed by the reference).
#define B_   128          // batch
#define C_   100          // channels
#define CP   112          // channels padded to 7 tiles of 16
#define N_   3136         // 56*56 positions
#define LDSTR 132         // padded batch stride in LDS floats:
                          //  - rows stay 16B-aligned (132*4 = 528 = 33*16)
                          //  - lane-to-lane bank step = 132 mod 64 = 4 -> 32 lanes
                          //    of an A/B fragment hit 32 distinct banks

static __device__ __constant__ float kScale = 0.08873565094f;  // 1/sqrt(127) -> Gram = cov/(B-1)
static __device__ __constant__ float kEps   = 0.01001f;        // 0.01 + 1e-5

typedef __attribute__((ext_vector_type(2))) float v2f;
typedef __attribute__((ext_vector_type(8))) float v8f;

// ---------------------------------------------------------------------------
// Kernel 1: mean over batch.  mean[c*N + n] = (1/B) sum_b emb[b,c,n]
// Consecutive threads -> consecutive n: fully coalesced.
// ---------------------------------------------------------------------------
__global__ void mvg_mean_kernel(const float* __restrict__ emb,
                                float* __restrict__ mean) {
  int idx = blockIdx.x * blockDim.x + threadIdx.x;
  if (idx >= C_ * N_) return;
  float s = 0.0f;
  for (int b = 0; b < B_; ++b)
    s += emb[(size_t)b * (C_ * N_) + idx];
  mean[idx] = s * (1.0f / B_);
}

// ---------------------------------------------------------------------------
// Kernel 2: tiled transpose + center + scale:
//   Y[n, c, b] = (emb[b,c,n] - mean[c,n]) * (1/sqrt(127))
// 32x32 (n x b) tile per block via LDS; coalesced reads (along n) and
// coalesced writes (along b).
// ---------------------------------------------------------------------------
__global__ void mvg_transpose_kernel(const float* __restrict__ emb,
                                     const float* __restrict__ mean,
                                     float* __restrict__ Y) {
  __shared__ float t[32][33];
  const int n0 = blockIdx.x * 32;
  const int b0 = blockIdx.y * 32;
  const int c  = blockIdx.z;
  const int tx = threadIdx.x, ty = threadIdx.y;

  for (int r = ty; r < 32; r += 8) {
    int b = b0 + r, n = n0 + tx;
    t[r][tx] = (emb[(size_t)b * (C_ * N_) + (size_t)c * N_ + n]
                - mean[c * N_ + n]) * kScale;
  }
  __syncthreads();
  for (int r = ty; r < 32; r += 8) {
    int n = n0 + r, b = b0 + tx;
    Y[(size_t)n * (C_ * B_) + c * B_ + b] = t[tx][r];  // (b=b0+tx, n=n0+r)
  }
}

// ---------------------------------------------------------------------------
// Kernel 3: per-position Gram matrix via V_WMMA_F32_16X16X4_F32.
// One workgroup (8 wave32) per position n. Stage centered/scaled X^T[c][b]
// into padded LDS via GLOBAL_LOAD_ASYNC_TO_LDS_B128 (ASYNCcnt path, bypasses
// VGPRs), then 49 = 7x7 output tiles of 16x16 with K=128 (32 x K=4 WMMA
// steps). Writes cov + eps*I into the inv_covariance output region.
//
// f32 WMMA fragment layouts (ISA 7.12.2):
//   A 16x4:  lane<16: {K0,K1}, lane>=16: {K2,K3}, M = lane%16
//   B 4x16:  lane<16: {K0,K1}, lane>=16: {K2,K3}, N = lane%16
//   C/D:     VGPR r, lane L -> (M = r + 8*(L>=16), N = L%16)
// ---------------------------------------------------------------------------
__global__ void mvg_cov_wmma_kernel(const float* __restrict__ Y,
                                    const float* __restrict__ emb,
                                    const float* __restrict__ mean,
                                    float* __restrict__ cov,
                                    int use_transposed) {
  __shared__ __align__(16) float xt[CP * LDSTR];  // 112 * 132 * 4 = 59,136 B
  const int n   = blockIdx.x;
  const int tid = threadIdx.x;

  if (use_transposed) {
    // Contiguous 51.2 KB per position. Async copy global -> LDS in 16-byte
    // chunks (each chunk lands inside one padded row; rows are 16B-aligned).
    const char* ysrc = (const char*)(Y + (size_t)n * (C_ * B_));
    for (int q = tid; q < CP * (B_ / 4); q += 256) {
      int c  = q / (B_ / 4);
      int q4 = q % (B_ / 4);
      float* dst = &xt[c * LDSTR + q4 * 4];
      if (c < C_) {
        // Flat->LDS aperture maps by 32-bit truncation (ISA 10.2).
        unsigned lds_addr = (unsigned)(uintptr_t)dst;
        unsigned long long gaddr =
            (unsigned long long)(uintptr_t)(ysrc + ((size_t)c * B_ + q4 * 4) * 4);
        asm volatile("global_load_async_to_lds_b128 %0, %1, off"
                     :: "v"(lds_addr), "v"(gaddr)
                     : "memory");
      } else {
        dst[0] = 0.f; dst[1] = 0.f; dst[2] = 0.f; dst[3] = 0.f;
      }
    }
    asm volatile("s_wait_asynccnt 0x0" ::: "memory");
  } else {
    // Fallback (no workspace): strided gather with on-the-fly centering.
    for (int q = tid; q < CP * B_; q += 256) {
      int c = q / B_, b = q % B_;
      float v = 0.0f;
      if (c < C_)
        v = (emb[(size_t)b * (C_ * N_) + (size_t)c * N_ + n]
             - mean[c * N_ + n]) * kScale;
      xt[c * LDSTR + b] = v;
    }
  }
  __syncthreads();

  const int wave = tid >> 5;
  const int lane = tid & 31;
  const int half = lane >> 4;   // 0: K-pair {0,1}; 1: K-pair {2,3}
  const int l    = lane & 15;

  float* out = cov + (size_t)n * (C_ * C_);

  for (int t = wave; t < 49; t += 8) {
    const int c0 = (t / 7) * 16;   // row tile (c)
    const int c1 = (t % 7) * 16;   // col tile (d)
    v8f acc = {0.f, 0.f, 0.f, 0.f, 0.f, 0.f, 0.f, 0.f};

    const float* arow = &xt[(c0 + l) * LDSTR + 2 * half];
    const float* brow = &xt[(c1 + l) * LDSTR + 2 * half];

    for (int kb = 0; kb < B_; kb += 4) {
      v2f a, b;
      a[0] = arow[kb];     a[1] = arow[kb + 1];
      b[0] = brow[kb];     b[1] = brow[kb + 1];
      acc = __builtin_amdgcn_wmma_f32_16x16x4_f32(
          /*neg_a=*/false, a, /*neg_b=*/false, b,
          /*c_mod=*/(short)0, acc, /*reuse_a=*/false, /*reuse_b=*/false);
    }

    for (int r = 0; r < 8; ++r) {
      int row = c0 + r + 8 * half;
      int col = c1 + l;
      if (row < C_ && col < C_) {
        float v = acc[r];
        if (row == col) v += kEps;
        out[row * C_ + col] = v;
      }
    }
  }
}

// ---------------------------------------------------------------------------
// Kernel 4: batched in-place Gauss-Jordan inverse (SPD + eps*I, no pivoting).
// One workgroup per position; matrix lives in padded LDS (100 x 101).
// ---------------------------------------------------------------------------
__global__ void mvg_invert_kernel(float* __restrict__ mats) {
  __shared__ float a[C_ * (C_ + 1)];   // 40,400 bytes
  __shared__ float fcol[C_];
  const int tid = threadIdx.x;
  float* M = mats + (size_t)blockIdx.x * (C_ * C_);

  for (int idx = tid; idx < C_ * C_; idx += 256)
    a[(idx / C_) * (C_ + 1) + (idx % C_)] = M[idx];
  __syncthreads();

  for (int k = 0; k < C_; ++k) {
    float piv = a[k * (C_ + 1) + k];
    if (tid < C_) fcol[tid] = a[tid * (C_ + 1) + k];
    __syncthreads();

    const float ip = 1.0f / piv;
    if (tid < C_)
      a[k * (C_ + 1) + tid] = (tid == k) ? ip : a[k * (C_ + 1) + tid] * ip;
    __syncthreads();

    for (int idx = tid; idx < C_ * C_; idx += 256) {
      int i = idx / C_;
      if (i == k) continue;
      int j = idx % C_;
      float f = fcol[i];
      float v = (j == k) ? (-f * ip)
                         : fmaf(-f, a[k * (C_ + 1) + j], a[i * (C_ + 1) + j]);
      a[i * (C_ + 1) + j] = v;
    }
    __syncthreads();
  }

  for (int idx = tid; idx < C_ * C_; idx += 256)
    M[idx] = a[(idx / C_) * (C_ + 1) + (idx % C_)];
}

// ---------------------------------------------------------------------------
extern "C" void kernel_launch(void* const* d_in, const int* in_sizes, int n_in,
                              void* d_out, int out_size, void* d_ws, size_t ws_size,
                              hipStream_t stream) {
  const float* emb = (const float*)d_in[0];
  float* out  = (float*)d_out;
  float* mean = out;                            // C*N floats
  float* inv  = out + (size_t)C_ * N_;          // N*C*C floats (cov -> inverse in place)
  float* Y    = (float*)d_ws;

  const size_t y_bytes = (size_t)N_ * C_ * B_ * sizeof(float);  // ~161 MB
  const int use_t = (d_ws != nullptr && ws_size >= y_bytes) ? 1 : 0;

  mvg_mean_kernel<<<(C_ * N_ + 255) / 256, 256, 0, stream>>>(emb, mean);

  if (use_t) {
    dim3 grid(N_ / 32, B_ / 32, C_);   // 98 x 4 x 100
    dim3 block(32, 8);
    mvg_transpose_kernel<<<grid, block, 0, stream>>>(emb, mean, Y);
  }

  mvg_cov_wmma_kernel<<<N_, 256, 0, stream>>>(use_t ? Y : nullptr,
                                              emb, mean, inv, use_t);

  mvg_invert_kernel<<<N_, 256, 0, stream>>>(inv);
}